// SigLayer_87333864997626
// MI455X (gfx1250) — compile-verified
//
#include <hip/hip_runtime.h>

// Truncated path signature (m=4, d=10, L=256) via Chen's identity.
// One workgroup (8 wave32) per batch; all state in LDS.
//
// Level-4 update as K=12 GEMM, 3 chained V_WMMA_F32_16X16X4_F32 per 16-row tile:
//   C(1000x10) += [S2 | seg2 | kron(S1)] x [v ; v/4 ; seg1(10x10)]
//   (kron column k' of row ijk holds S1[ij]*delta_{k,k'})
// leaving only S0(x)seg2 for the VALU writeback.
// Level-3 update likewise: C(100x10) += [S1 | seg1 | kron(S0)] x [v ; v/3 ; seg1],
// 7 tiles on waves 0..6, writeback is a pure accumulate.
// All lane-half-dependent kron constants / match booleans are hoisted out of
// the 255-step scan so A builds are pure cmp+cndmask (no exec-mask branches).

typedef __attribute__((ext_vector_type(2))) float v2f;
typedef __attribute__((ext_vector_type(8))) float v8f;

#define DD     10
#define LPATH  256
#define NSTEP  (LPATH - 1)

#define WMMA4(Amat, Bmat, Cacc)                                        \
  __builtin_amdgcn_wmma_f32_16x16x4_f32(false, (Amat), false, (Bmat),  \
                                        (short)0, (Cacc), false, false)

__global__ __launch_bounds__(256) void sig4_kernel(const float* __restrict__ x,
                                                   float* __restrict__ out) {
  __shared__ float xb[LPATH * DD];  // staged path
  __shared__ float vp[16];          // increment, zero-padded
  __shared__ float seg1[128];       // v_i v_j / 2        (100 used, zero pad)
  __shared__ float seg2p[1024];     // v_i v_j v_k / 6    (1000 used, zero pad)
  __shared__ float S0[16];          // level 1            (10 used, zero pad)
  __shared__ float S1[128];         // level 2            (100 used, zero pad)
  __shared__ float S2p[1120];       // level 3            (1000 used, zero pad)
  __shared__ float S3p[10080];      // level 4            (10000 used, pad rows)
  __shared__ int   tab[1008];       // packed {i, ij, jk, j, k} per row r

  const int b    = blockIdx.x;
  const int tid  = threadIdx.x;
  const int lane = tid & 31;
  const int wave = tid >> 5;
  const int lm   = lane & 15;
  const bool hi  = lane >= 16;
  const int n    = lm;              // WMMA column owned by this lane

  // ---- stage path, zero state (incl. padding), build index table ----
  const float* xp = x + (size_t)b * (LPATH * DD);
  for (int e = tid; e < LPATH * DD; e += 256) xb[e] = xp[e];
  if (tid < 16) { vp[tid] = 0.0f; S0[tid] = 0.0f; }
  if (tid < 128) { seg1[tid] = 0.0f; S1[tid] = 0.0f; }
  for (int e = tid; e < 1024; e += 256) seg2p[e] = 0.0f;
  for (int e = tid; e < 1120; e += 256) S2p[e] = 0.0f;
  for (int e = tid; e < 10080; e += 256) S3p[e] = 0.0f;
  for (int r = tid; r < 1008; r += 256) {
    int p = 0;
    if (r < 1000) {
      const int i = r / 100, rem = r % 100, j = rem / 10, k = rem % 10;
      p = i | ((i * 10 + j) << 4) | ((j * 10 + k) << 11) | (j << 18) | (k << 22);
    }
    tab[r] = p;
  }
  // t-invariant per-lane constants (hoisted out of the scan):
  const int i2 = tid / 10, j2 = tid % 10;      // level-2 decomposition (tid<100)
  const int r3 = wave * 16 + lm;               // level-3 tile row (waves 0..6)
  const int i3 = r3 / 10, j3 = r3 % 10;        // r3 <= 111 -> i3 <= 11 < 16
  // level-4 kron comparison constants (per lane-half):
  const int kc0x = hi ? 0 : 99;                // A0.x hi slot: k'=0 (lo: never)
  const int kc0y = hi ? 1 : 99;                // A0.y hi slot: k'=1
  const int kc1x = hi ? 4 : 2;                 // A1.x: k'=4|2
  const int kc1y = hi ? 5 : 3;                 // A1.y: k'=5|3
  const int kc2x = hi ? 8 : 6;                 // A2.x: k'=8|6
  const int kc2y = hi ? 9 : 7;                 // A2.y: k'=9|7
  // level-3 kron match booleans (fully t-invariant):
  const bool p0x = hi && (j3 == 0), p0y = hi && (j3 == 1);
  const bool p1x = (j3 == kc1x),    p1y = (j3 == kc1y);
  const bool p2x = (j3 == kc2x),    p2y = (j3 == kc2y);
  __syncthreads();

  for (int t = 0; t < NSTEP; ++t) {
    // ---- increment v = x[t+1] - x[t] ----
    if (tid < DD) vp[tid] = xb[(t + 1) * DD + tid] - xb[t * DD + tid];
    __syncthreads();

    // ---- segment levels: seg1 = v(x)v/2, seg2 = v(x)v(x)v/6 ----
    if (tid < 100) seg1[tid] = vp[i2] * vp[j2] * 0.5f;
    for (int e = tid; e < 1000; e += 256) {
      const int p = tab[e];
      seg2p[e] = vp[p & 15] * vp[(p >> 18) & 15] * vp[(p >> 22) & 15] * (1.0f / 6.0f);
    }
    __syncthreads();

    // ---- build shared B chunks (per wave, per step) ----
    // B 4x16 layout: VGPR0 = {row 4c | row 4c+2}, VGPR1 = {row 4c+1 | row 4c+3}.
    const float vn  = vp[n];                   // n<16, padded
    const float g0n = seg1[n],      g1n = seg1[10 + n];
    const float g2n = seg1[20 + n], g3n = seg1[30 + n];
    const float g4n = seg1[40 + n], g5n = seg1[50 + n];
    const float g6n = seg1[60 + n], g7n = seg1[70 + n];
    const float g8n = seg1[80 + n], g9n = seg1[90 + n];
    v2f B0_4, B0_3, B1, B2;
    B0_4.x = hi ? g0n : vn;           B0_4.y = hi ? g1n : vn * 0.25f;
    B0_3.x = hi ? g0n : vn;           B0_3.y = hi ? g1n : vn * (1.0f / 3.0f);
    B1.x   = hi ? g4n : g2n;          B1.y   = hi ? g5n : g3n;
    B2.x   = hi ? g8n : g6n;          B2.y   = hi ? g9n : g7n;

    // ---- level 4: S3 += S2(x)v + seg3 + S1(x)seg1 (WMMA) + S0(x)seg2 (VALU) ----
    for (int tile = wave; tile < 63; tile += 8) {   // wave-uniform trip count
      const int r0 = tile * 16;
      const int ra = r0 + lm;                  // <= 1007: padded-safe
      const int pA = tab[ra];
      const int kA = (pA >> 22) & 15;
      const float s2v = S2p[ra];
      const float sgv = seg2p[ra];
      const float s1v = S1[(pA >> 4) & 127];
      v2f A0, A1, A2;                          // A builds: cmp + cndmask only
      const float h0x = (kA == kc0x) ? s1v : 0.0f;
      const float h0y = (kA == kc0y) ? s1v : 0.0f;
      A0.x = hi ? h0x : s2v;
      A0.y = hi ? h0y : sgv;
      A1.x = (kA == kc1x) ? s1v : 0.0f;
      A1.y = (kA == kc1y) ? s1v : 0.0f;
      A2.x = (kA == kc2x) ? s1v : 0.0f;
      A2.y = (kA == kc2y) ? s1v : 0.0f;
      v8f acc = {};
      acc = WMMA4(A0, B0_4, acc);
      acc = WMMA4(A1, B1, acc);
      acc = WMMA4(A2, B2, acc);

      if (n < DD) {                            // D: VGPR g -> (M=g | M=g+8, N=lm)
        const int rbase = r0 + (hi ? 8 : 0);
#pragma unroll
        for (int g = 0; g < 8; ++g) {
          const int r = rbase + g;             // rows >= 1000 hit padding
          const int p = tab[r];
          const float add = acc[g] + S0[p & 15] * seg2p[((p >> 11) & 127) * 10 + n];
          S3p[r * 10 + n] += add;
        }
      }
    }
    __syncthreads();

    // ---- level 3: S2 += S1(x)v + seg2 + S0(x)seg1 — all WMMA, waves 0..6 ----
    if (wave < 7) {
      const float s1r  = S1[r3];               // r3 <= 111 < 128 (pad zeros)
      const float sg1r = seg1[r3];
      const float s0v  = S0[i3];               // i3 <= 11 < 16 (pad zeros)
      v2f A0, A1, A2;
      A0.x = hi ? (p0x ? s0v : 0.0f) : s1r;
      A0.y = hi ? (p0y ? s0v : 0.0f) : sg1r;
      A1.x = p1x ? s0v : 0.0f;
      A1.y = p1y ? s0v : 0.0f;
      A2.x = p2x ? s0v : 0.0f;
      A2.y = p2y ? s0v : 0.0f;
      v8f acc = {};
      acc = WMMA4(A0, B0_3, acc);
      acc = WMMA4(A1, B1, acc);
      acc = WMMA4(A2, B2, acc);

      if (n < DD) {
        const int rbase = wave * 16 + (hi ? 8 : 0);
#pragma unroll
        for (int g = 0; g < 8; ++g) {
          S2p[(rbase + g) * 10 + n] += acc[g]; // rows >= 100 accumulate exact 0
        }
      }
    }
    __syncthreads();

    // ---- level 2: S1 += seg1 + S0(x)v ----
    if (tid < 100) S1[tid] += seg1[tid] + S0[i2] * vp[j2];
    __syncthreads();

    // ---- level 1: S0 += v ----
    if (tid < DD) S0[tid] += vp[tid];
    __syncthreads();
  }

  // ---- write concatenated levels 1..4 (reference layout) ----
  float* op = out + (size_t)b * 11110;
  for (int e = tid; e < 10; e += 256) op[e] = S0[e];
  for (int e = tid; e < 100; e += 256) op[10 + e] = S1[e];
  for (int e = tid; e < 1000; e += 256) op[110 + e] = S2p[e];
  for (int e = tid; e < 10000; e += 256) op[1110 + e] = S3p[e];
}

extern "C" void kernel_launch(void* const* d_in, const int* in_sizes, int n_in,
                              void* d_out, int out_size, void* d_ws, size_t ws_size,
                              hipStream_t stream) {
  const float* x = (const float*)d_in[0];
  float* out = (float*)d_out;
  const int B = in_sizes[0] / (LPATH * DD);  // 2048
  sig4_kernel<<<B, 256, 0, stream>>>(x, out);
}